// GNN_6339371728976
// MI455X (gfx1250) — compile-verified
//
#include <hip/hip_runtime.h>
#include <hip/hip_bf16.h>

// ---------------------------------------------------------------------------
// GIN (eps=0) x3 layers + mean pool + linear head, for MI455X (gfx1250).
// GEMMs use V_WMMA_F32_16X16X4_F32 (full f32, matches reference numerics).
// Scatter/pool use hardware f32 global atomics (L2-resident working set).
// ---------------------------------------------------------------------------

typedef __attribute__((ext_vector_type(2))) float v2f;
typedef __attribute__((ext_vector_type(8))) float v8f;

#define GNN_N   50000
#define GNN_E   800000
#define GNN_D   64
#define GNN_L   3
#define GNN_G   128
#define GNN_OUT 32

// ---------------------------------------------------------------- zero fill
__global__ void gnn_zero_kernel(float* __restrict__ p, int n) {
    int t = blockIdx.x * blockDim.x + threadIdx.x;
    if (t < n) p[t] = 0.0f;
}

// ------------------------------------------------------- edge scatter (agg)
// agg[dst] += h[src];  16 threads per edge, float4 gather + 4 f32 atomics.
__global__ void gnn_scatter_kernel(const float* __restrict__ h,
                                   const int* __restrict__ src,
                                   const int* __restrict__ dst,
                                   float* __restrict__ agg) {
    int t = blockIdx.x * blockDim.x + threadIdx.x;
    int e = t >> 4;
    if (e >= GNN_E) return;
    int f = (t & 15) * 4;
    int s = src[e];
    int d = dst[e];
    const float4 v = *reinterpret_cast<const float4*>(h + (size_t)s * GNN_D + f);
    float* o = agg + (size_t)d * GNN_D + f;
    unsafeAtomicAdd(o + 0, v.x);
    unsafeAtomicAdd(o + 1, v.y);
    unsafeAtomicAdd(o + 2, v.z);
    unsafeAtomicAdd(o + 3, v.w);
}

// ------------------------------------------------- fused GIN MLP (WMMA f32)
// hout = relu( relu( (agg+h) @ W1 + b1 ) @ W2 + b2 )
// 128 threads = 4 waves; block owns 16 rows; wave w owns output cols
// [16w,16w+16). K=64 swept as 16 WMMA_F32_16X16X4_F32 steps.
__global__ __launch_bounds__(128)
void gnn_mlp_kernel(const float* __restrict__ h,
                    const float* __restrict__ agg,
                    const float* __restrict__ W1,
                    const float* __restrict__ b1,
                    const float* __restrict__ W2,
                    const float* __restrict__ b2,
                    float* __restrict__ hout) {
    __shared__ float Atile[16][GNN_D + 1];   // +1 pad: kills bank conflicts
    __shared__ float Hid[16][GNN_D + 1];

    const int row0 = blockIdx.x * 16;
    const int tid  = threadIdx.x;
    const int wave = tid >> 5;          // 0..3 -> output column tile
    const int lane = tid & 31;
    const int am   = lane & 15;         // A-matrix row within tile
    const int hi   = lane >> 4;         // 0: K pair {4k,4k+1}; 1: {4k+2,4k+3}
    const int col  = wave * 16 + am;    // output column (B/C/D N index)

    // Stage A = agg + h into LDS (16x64)
    for (int i = tid; i < 16 * GNN_D; i += 128) {
        int r = i >> 6, c = i & 63;
        size_t g = (size_t)(row0 + r) * GNN_D + c;
        Atile[r][c] = agg[g] + h[g];
    }
    __syncthreads();

    // ---- GEMM 1: (A @ W1 + b1), relu, -> LDS Hid -------------------------
    v8f acc;
    {
        float bias = b1[col];
        for (int r = 0; r < 8; ++r) acc[r] = bias;
    }
    for (int k = 0; k < 16; ++k) {
        const int kb = 4 * k + 2 * hi;
        v2f a, b;
        a.x = Atile[am][kb];
        a.y = Atile[am][kb + 1];
        b.x = W1[(size_t)kb * GNN_D + col];
        b.y = W1[(size_t)(kb + 1) * GNN_D + col];
        acc = __builtin_amdgcn_wmma_f32_16x16x4_f32(false, a, false, b,
                                                    (short)0, acc, false, false);
    }
    for (int r = 0; r < 8; ++r) {
        float v = acc[r];
        Hid[hi * 8 + r][col] = v > 0.0f ? v : 0.0f;
    }
    __syncthreads();

    // ---- GEMM 2: (Hid @ W2 + b2), relu, -> global ------------------------
    v8f acc2;
    {
        float bias = b2[col];
        for (int r = 0; r < 8; ++r) acc2[r] = bias;
    }
    for (int k = 0; k < 16; ++k) {
        const int kb = 4 * k + 2 * hi;
        v2f a, b;
        a.x = Hid[am][kb];
        a.y = Hid[am][kb + 1];
        b.x = W2[(size_t)kb * GNN_D + col];
        b.y = W2[(size_t)(kb + 1) * GNN_D + col];
        acc2 = __builtin_amdgcn_wmma_f32_16x16x4_f32(false, a, false, b,
                                                     (short)0, acc2, false, false);
    }
    for (int r = 0; r < 8; ++r) {
        float v = acc2[r];
        hout[(size_t)(row0 + hi * 8 + r) * GNN_D + col] = v > 0.0f ? v : 0.0f;
    }
}

// ------------------------------------------------------- mean-pool accumulate
__global__ void gnn_pool_kernel(const float* __restrict__ h,
                                const int* __restrict__ batch,
                                float* __restrict__ sums,
                                float* __restrict__ cnts) {
    int t = blockIdx.x * blockDim.x + threadIdx.x;
    int n = t >> 4;
    if (n >= GNN_N) return;
    int part = t & 15;
    int f = part * 4;
    int g = batch[n];
    const float4 v = *reinterpret_cast<const float4*>(h + (size_t)n * GNN_D + f);
    float* o = sums + (size_t)g * GNN_D + f;
    unsafeAtomicAdd(o + 0, v.x);
    unsafeAtomicAdd(o + 1, v.y);
    unsafeAtomicAdd(o + 2, v.z);
    unsafeAtomicAdd(o + 3, v.w);
    if (part == 0) unsafeAtomicAdd(&cnts[g], 1.0f);
}

__global__ void gnn_mean_kernel(const float* __restrict__ sums,
                                const float* __restrict__ cnts,
                                float* __restrict__ mean) {
    int t = blockIdx.x * blockDim.x + threadIdx.x;   // G*D threads
    if (t >= GNN_G * GNN_D) return;
    float c = cnts[t >> 6];
    c = c < 1.0f ? 1.0f : c;
    mean[t] = sums[t] / c;
}

// ----------------------------------------------------- head GEMM (WMMA f32)
// out[128x32] = mean[128x64] @ lin_w[64x32] + lin_b.  2 waves/block, 8 blocks.
__global__ __launch_bounds__(64)
void gnn_head_kernel(const float* __restrict__ mean,
                     const float* __restrict__ lw,
                     const float* __restrict__ lb,
                     float* __restrict__ out) {
    const int row0 = blockIdx.x * 16;
    const int wave = threadIdx.x >> 5;   // 0..1 -> cols [0,16) / [16,32)
    const int lane = threadIdx.x & 31;
    const int am   = lane & 15;
    const int hi   = lane >> 4;
    const int col  = wave * 16 + am;

    v8f acc;
    {
        float bias = lb[col];
        for (int r = 0; r < 8; ++r) acc[r] = bias;
    }
    for (int k = 0; k < 16; ++k) {
        const int kb = 4 * k + 2 * hi;
        v2f a, b;
        a.x = mean[(size_t)(row0 + am) * GNN_D + kb];
        a.y = mean[(size_t)(row0 + am) * GNN_D + kb + 1];
        b.x = lw[(size_t)kb * GNN_OUT + col];
        b.y = lw[(size_t)(kb + 1) * GNN_OUT + col];
        acc = __builtin_amdgcn_wmma_f32_16x16x4_f32(false, a, false, b,
                                                    (short)0, acc, false, false);
    }
    for (int r = 0; r < 8; ++r)
        out[(size_t)(row0 + hi * 8 + r) * GNN_OUT + col] = acc[r];
}

// ---------------------------------------------------------------------------
extern "C" void kernel_launch(void* const* d_in, const int* in_sizes, int n_in,
                              void* d_out, int out_size, void* d_ws, size_t ws_size,
                              hipStream_t stream) {
    const float* x     = (const float*)d_in[0];
    const int*   ei    = (const int*)d_in[1];   // (2,E) row-major
    const int*   batch = (const int*)d_in[2];
    const float* W1    = (const float*)d_in[3];
    const float* b1    = (const float*)d_in[4];
    const float* W2    = (const float*)d_in[5];
    const float* b2    = (const float*)d_in[6];
    const float* lw    = (const float*)d_in[7];
    const float* lb    = (const float*)d_in[8];
    float*       out   = (float*)d_out;

    const int* src = ei;
    const int* dst = ei + GNN_E;

    // Workspace layout (f32): agg | hA | hB | sums | cnts | mean
    float* ws   = (float*)d_ws;
    float* agg  = ws;
    float* hA   = agg + (size_t)GNN_N * GNN_D;
    float* hB   = hA  + (size_t)GNN_N * GNN_D;
    float* sums = hB  + (size_t)GNN_N * GNN_D;
    float* cnts = sums + (size_t)GNN_G * GNN_D;
    float* mean = cnts + GNN_G;

    const int ND = GNN_N * GNN_D;

    const float* hin = x;
    float* cur = hA;
    float* nxt = hB;

    for (int l = 0; l < GNN_L; ++l) {
        gnn_zero_kernel<<<(ND + 255) / 256, 256, 0, stream>>>(agg, ND);
        gnn_scatter_kernel<<<(GNN_E * 16 + 255) / 256, 256, 0, stream>>>(
            hin, src, dst, agg);
        gnn_mlp_kernel<<<GNN_N / 16, 128, 0, stream>>>(
            hin, agg,
            W1 + (size_t)l * GNN_D * GNN_D, b1 + (size_t)l * GNN_D,
            W2 + (size_t)l * GNN_D * GNN_D, b2 + (size_t)l * GNN_D,
            cur);
        hin = cur;
        float* tmp = cur; cur = nxt; nxt = tmp;
    }
    // hin now points at the final node features.

    const int PC = GNN_G * GNN_D + GNN_G;     // sums + cnts (contiguous)
    gnn_zero_kernel<<<(PC + 255) / 256, 256, 0, stream>>>(sums, PC);
    gnn_pool_kernel<<<(GNN_N * 16 + 255) / 256, 256, 0, stream>>>(
        hin, batch, sums, cnts);
    gnn_mean_kernel<<<(GNN_G * GNN_D + 255) / 256, 256, 0, stream>>>(
        sums, cnts, mean);
    gnn_head_kernel<<<GNN_G / 16, 64, 0, stream>>>(mean, lw, lb, out);
}